// SelectiveSSM_18794776887746
// MI455X (gfx1250) — compile-verified
//
#include <hip/hip_runtime.h>

typedef float v2f __attribute__((ext_vector_type(2)));
typedef float v8f __attribute__((ext_vector_type(8)));

#define B_SZ 4
#define L_SZ 2048
#define D_SZ 1024
#define N_SZ 16
#define R_SZ 64
#define E_SZ 96                 // R + 2N
#define CH   64                 // chunk length for associative scan
#define NCH  (L_SZ / CH)        // 32 chunks
#define M_SZ (B_SZ * L_SZ)      // 8192 rows

// workspace offsets (floats)
#define WS_XDBL  0
#define WS_DELTA (WS_XDBL  + (size_t)M_SZ * E_SZ)               // 786432
#define WS_S     (WS_DELTA + (size_t)M_SZ * D_SZ)               // + 8388608
#define WS_SUMD  (WS_S     + (size_t)B_SZ * NCH * D_SZ * N_SZ)  // + 2097152
#define WS_HIN   (WS_SUMD  + (size_t)B_SZ * NCH * D_SZ)         // + 131072
// total = 13,500,416 floats = ~52 MB

// ---------------------------------------------------------------------------
// K1: x_dbl = X (8192x1024) * W_xproj^T (1024x96), fp32 WMMA 16x16x4.
// One wave per 16x96 row panel (6 accumulators), K-loop of 256 steps.
// ---------------------------------------------------------------------------
__global__ void ssm_k1_xproj(const float* __restrict__ X,
                             const float* __restrict__ W,
                             float* __restrict__ xdbl) {
  const int lane = threadIdx.x & 31;
  const int wave = threadIdx.x >> 5;
  const int mt = blockIdx.x * 8 + wave;          // 0..511 M-tiles
  const int kh = lane >> 4;                       // half-wave selects K pair
  const int ln = lane & 15;

  const float* arow = X + (size_t)(mt * 16 + ln) * D_SZ + 2 * kh;
  v8f acc[6];
#pragma unroll
  for (int nt = 0; nt < 6; ++nt) acc[nt] = {};

  for (int ks = 0; ks < D_SZ / 4; ++ks) {
    const int kb = ks * 4;
    v2f a = *(const v2f*)(arow + kb);
#pragma unroll
    for (int nt = 0; nt < 6; ++nt) {
      v2f bv = *(const v2f*)(W + (size_t)(nt * 16 + ln) * D_SZ + kb + 2 * kh);
      acc[nt] = __builtin_amdgcn_wmma_f32_16x16x4_f32(
          false, a, false, bv, (short)0, acc[nt], false, false);
    }
  }
#pragma unroll
  for (int nt = 0; nt < 6; ++nt) {
#pragma unroll
    for (int v = 0; v < 8; ++v) {
      xdbl[(size_t)(mt * 16 + v + 8 * kh) * E_SZ + nt * 16 + ln] = acc[nt][v];
    }
  }
}

// ---------------------------------------------------------------------------
// K2: delta = softplus(x_dbl[:, :64] * W_dt^T + b_dt), fp32 WMMA 16x16x4.
// One wave per 16x16 output tile; 32768 waves total.
// ---------------------------------------------------------------------------
__global__ void ssm_k2_delta(const float* __restrict__ xdbl,
                             const float* __restrict__ Wdt,
                             const float* __restrict__ bdt,
                             float* __restrict__ delta) {
  const int lane = threadIdx.x & 31;
  const int wave = threadIdx.x >> 5;
  const int w = blockIdx.x * 8 + wave;            // 0..32767
  const int mt = w >> 6;                           // 512 M-tiles
  const int dt = w & 63;                           // 64 N-tiles over D
  const int kh = lane >> 4;
  const int ln = lane & 15;

  const float* arow = xdbl + (size_t)(mt * 16 + ln) * E_SZ + 2 * kh;
  const float* brow = Wdt  + (size_t)(dt * 16 + ln) * R_SZ + 2 * kh;
  v8f acc = {};
#pragma unroll
  for (int ks = 0; ks < R_SZ / 4; ++ks) {
    v2f a = *(const v2f*)(arow + ks * 4);
    v2f b = *(const v2f*)(brow + ks * 4);
    acc = __builtin_amdgcn_wmma_f32_16x16x4_f32(
        false, a, false, b, (short)0, acc, false, false);
  }
  const int d = dt * 16 + ln;
  const float bias = bdt[d];
#pragma unroll
  for (int v = 0; v < 8; ++v) {
    const int m = mt * 16 + v + 8 * kh;
    const float z = acc[v] + bias;
    // numerically-stable softplus
    const float sp = (z > 0.f) ? z + log1pf(__expf(-z)) : log1pf(__expf(z));
    delta[(size_t)m * D_SZ + d] = sp;
  }
}

// ---------------------------------------------------------------------------
// K3: per-chunk local scan (h starts at 0). Emits chunk-final state S and
// chunk sum of delta (cumulative dA product = exp(a * sum_delta)).
// grid = (D/256, NCH, B), block = 256 (one d per thread).
// ---------------------------------------------------------------------------
__global__ void ssm_k3_chunk(const float* __restrict__ delta,
                             const float* __restrict__ x,
                             const float* __restrict__ xdbl,
                             const float* __restrict__ A_log,
                             float* __restrict__ S,
                             float* __restrict__ sumd) {
  __shared__ float sB[CH * N_SZ];
  const int tid = threadIdx.x;
  const int d = blockIdx.x * 256 + tid;
  const int c = blockIdx.y;
  const int b = blockIdx.z;
  const int t0 = c * CH;

  for (int i = tid; i < CH * N_SZ; i += 256) {
    const int t = i >> 4, n = i & 15;
    sB[i] = xdbl[(size_t)(b * L_SZ + t0 + t) * E_SZ + R_SZ + n];
  }
  __syncthreads();

  float a[N_SZ], h[N_SZ];
#pragma unroll
  for (int n = 0; n < N_SZ; ++n) {
    a[n] = -__expf(A_log[d * N_SZ + n]);
    h[n] = 0.f;
  }
  float sd = 0.f;
  for (int t = 0; t < CH; ++t) {
    const size_t idx = (size_t)(b * L_SZ + t0 + t) * D_SZ + d;
    __builtin_prefetch(&delta[idx + 8 * D_SZ], 0, 1);
    const float dl = delta[idx];
    const float xv = x[idx];
    sd += dl;
    const float s = dl * xv;
#pragma unroll
    for (int n = 0; n < N_SZ; ++n) {
      h[n] = __expf(a[n] * dl) * h[n] + s * sB[t * N_SZ + n];
    }
  }
  const size_t base = (size_t)(b * NCH + c) * D_SZ + d;
#pragma unroll
  for (int n = 0; n < N_SZ; ++n) S[base * N_SZ + n] = h[n];
  sumd[base] = sd;
}

// ---------------------------------------------------------------------------
// K4: cross-chunk carry scan. One thread per (b,d,n); 32 sequential chunks.
// Hin[b,c,d,n] = state entering chunk c.
// ---------------------------------------------------------------------------
__global__ void ssm_k4_carry(const float* __restrict__ A_log,
                             const float* __restrict__ S,
                             const float* __restrict__ sumd,
                             float* __restrict__ Hin) {
  const int idx = blockIdx.x * 256 + threadIdx.x;   // B*D*N = 65536
  const int n = idx & 15;
  const int d = (idx >> 4) & (D_SZ - 1);
  const int b = idx >> 14;
  const float a = -__expf(A_log[d * N_SZ + n]);
  float h = 0.f;
  for (int c = 0; c < NCH; ++c) {
    const size_t base = (size_t)(b * NCH + c) * D_SZ + d;
    Hin[base * N_SZ + n] = h;
    h = __expf(a * sumd[base]) * h + S[base * N_SZ + n];
  }
}

// ---------------------------------------------------------------------------
// K5: final chunk scan with correct incoming state; y = sum_n h*C + D*x.
// ---------------------------------------------------------------------------
__global__ void ssm_k5_final(const float* __restrict__ delta,
                             const float* __restrict__ x,
                             const float* __restrict__ xdbl,
                             const float* __restrict__ A_log,
                             const float* __restrict__ Hin,
                             const float* __restrict__ Dp,
                             float* __restrict__ out) {
  __shared__ float sB[CH * N_SZ];
  __shared__ float sC[CH * N_SZ];
  const int tid = threadIdx.x;
  const int d = blockIdx.x * 256 + tid;
  const int c = blockIdx.y;
  const int b = blockIdx.z;
  const int t0 = c * CH;

  for (int i = tid; i < CH * N_SZ; i += 256) {
    const int t = i >> 4, n = i & 15;
    const size_t row = (size_t)(b * L_SZ + t0 + t) * E_SZ;
    sB[i] = xdbl[row + R_SZ + n];
    sC[i] = xdbl[row + R_SZ + N_SZ + n];
  }
  __syncthreads();

  float a[N_SZ], h[N_SZ];
  const size_t cbase = (size_t)(b * NCH + c) * D_SZ + d;
#pragma unroll
  for (int n = 0; n < N_SZ; ++n) {
    a[n] = -__expf(A_log[d * N_SZ + n]);
    h[n] = Hin[cbase * N_SZ + n];
  }
  const float dpv = Dp[d];
  for (int t = 0; t < CH; ++t) {
    const size_t idx = (size_t)(b * L_SZ + t0 + t) * D_SZ + d;
    __builtin_prefetch(&delta[idx + 8 * D_SZ], 0, 1);
    const float dl = delta[idx];
    const float xv = x[idx];
    const float s = dl * xv;
    float y = 0.f;
#pragma unroll
    for (int n = 0; n < N_SZ; ++n) {
      h[n] = __expf(a[n] * dl) * h[n] + s * sB[t * N_SZ + n];
      y += h[n] * sC[t * N_SZ + n];
    }
    out[idx] = y + dpv * xv;
  }
}

// ---------------------------------------------------------------------------
extern "C" void kernel_launch(void* const* d_in, const int* in_sizes, int n_in,
                              void* d_out, int out_size, void* d_ws, size_t ws_size,
                              hipStream_t stream) {
  (void)in_sizes; (void)n_in; (void)out_size; (void)ws_size;
  const float* x    = (const float*)d_in[0];
  const float* Wx   = (const float*)d_in[1];
  const float* Wdt  = (const float*)d_in[2];
  const float* bdt  = (const float*)d_in[3];
  const float* Alog = (const float*)d_in[4];
  const float* Dp   = (const float*)d_in[5];
  float* out = (float*)d_out;

  float* ws    = (float*)d_ws;
  float* xdbl  = ws + WS_XDBL;
  float* delta = ws + WS_DELTA;
  float* S     = ws + WS_S;
  float* sumd  = ws + WS_SUMD;
  float* Hin   = ws + WS_HIN;

  // 512 M-tiles, 8 waves per 256-thread block
  ssm_k1_xproj<<<64, 256, 0, stream>>>(x, Wx, xdbl);
  // 32768 tile-waves, 8 per block
  ssm_k2_delta<<<4096, 256, 0, stream>>>(xdbl, Wdt, bdt, delta);
  // chunked scan: phase 1 (local), phase 2 (carry), phase 3 (final)
  ssm_k3_chunk<<<dim3(D_SZ / 256, NCH, B_SZ), 256, 0, stream>>>(
      delta, x, xdbl, Alog, S, sumd);
  ssm_k4_carry<<<(B_SZ * D_SZ * N_SZ) / 256, 256, 0, stream>>>(
      Alog, S, sumd, Hin);
  ssm_k5_final<<<dim3(D_SZ / 256, NCH, B_SZ), 256, 0, stream>>>(
      delta, x, xdbl, Alog, Hin, Dp, out);
}